// BMA_update_10204842295849
// MI455X (gfx1250) — compile-verified
//
#include <hip/hip_runtime.h>

// ---------------- CDNA5 WMMA types ----------------
typedef __attribute__((ext_vector_type(2))) float v2f;
typedef __attribute__((ext_vector_type(8))) float v8f;

#define E_NUM   1600000
#define NT      8          // 128 output cols / 16
#define KS1     32         // K=128 -> 32 wmma k-steps of 4
#define KSR     16         // K=64  -> 16 wmma k-steps of 4
#define RS_PAIR 132        // row stride (dwords) for 128-wide A staging (+4 pad: bank-conflict-free)
#define RS_ATTR 68         // row stride (dwords) for 64-wide A staging  (+4 pad)

// LDS layout (dword offsets)
#define OFF_W1   0                        // packed W_nnn : 8*32*64 = 16384 dw (64 KB)
#define OFF_WR   (OFF_W1 + NT*KS1*64)     // packed W_root: 8*16*64 =  8192 dw (32 KB)
#define OFF_W2   (OFF_WR + NT*KSR*64)     // packed W_out : 8*32*64 = 16384 dw (64 KB)
#define OFF_WAVE (OFF_W2 + NT*KS1*64)     // per-wave staging starts at 40960 dw
#define WAVE_DW  (16*RS_PAIR + 16*RS_ATTR + 32)   // pairA + attrA + 32 edge indices = 3232 dw
#define TOTAL_DW (OFF_WAVE + 8*WAVE_DW)           // 66816 dw = 267264 B (<= 320KB WGP LDS)

// Pack weight matrix W[K x 128] (row-major) into B-fragment order:
// packed[((nt*ksteps + k)*32 + lane)*2 + j] = W[(4k + 2*(lane>>4) + j)*128 + nt*16 + (lane&15)]
// -> each WMMA B operand is one aligned, bank-conflict-free ds_load_b64 at an immediate offset.
__device__ __forceinline__ void pack_weights(const float* __restrict__ W, float* __restrict__ lds,
                                             int ksteps, int tid, int nthreads) {
    const int total = NT * ksteps * 64;
    for (int f = tid; f < total; f += nthreads) {
        const int j  = f & 1;
        const int ln = (f >> 1) & 31;
        const int k  = (f >> 6) % ksteps;
        const int nt = f / (ksteps * 64);
        const int kk = 4 * k + 2 * (ln >> 4) + j;
        lds[f] = W[kk * 128 + nt * 16 + (ln & 15)];
    }
}

extern "C" __global__ void __launch_bounds__(256)
edge_mlp_wmma_f32(const float* __restrict__ x,
                  const long long* __restrict__ eidx,
                  const float* __restrict__ eattr,
                  const float* __restrict__ Wnnn,  const float* __restrict__ bnnn,
                  const float* __restrict__ Wroot, const float* __restrict__ broot,
                  const float* __restrict__ Wout,  const float* __restrict__ bout,
                  float* __restrict__ out)
{
    extern __shared__ float smem[];
    const int tid  = threadIdx.x;
    const int lane = tid & 31;
    const int wave = tid >> 5;
    const int hi   = lane >> 4;   // half-wave select (K sub-pair / M upper rows)
    const int col  = lane & 15;   // N column within tile / M row for A

    // ---- stage all weights into LDS once per block (L2-resident source) ----
    pack_weights(Wnnn,  smem + OFF_W1, KS1, tid, 256);
    pack_weights(Wroot, smem + OFF_WR, KSR, tid, 256);
    pack_weights(Wout,  smem + OFF_W2, KS1, tid, 256);
    __syncthreads();

    float* pairA = smem + OFF_WAVE + wave * WAVE_DW;   // 16 x 128 (stride 132), reused as hiddenA
    float* attrA = pairA + 16 * RS_PAIR;               // 16 x 64  (stride 68)
    int*   idxB  = (int*)(attrA + 16 * RS_ATTR);       // 32 node indices (src[16] | dst[16])

    const v2f* bW1 = (const v2f*)(smem + OFF_W1) + lane;
    const v2f* bWR = (const v2f*)(smem + OFF_WR) + lane;
    const v2f* bW2 = (const v2f*)(smem + OFF_W2) + lane;

    // per-lane biases (L2/L1-resident, loaded once)
    float bb_n[NT], bb_r[NT], bb_o[NT];
    #pragma unroll
    for (int nt = 0; nt < NT; ++nt) {
        bb_n[nt] = bnnn[nt * 16 + col];
        bb_r[nt] = broot[nt * 16 + col];
        bb_o[nt] = bout[nt * 16 + col];
    }

    const int numTiles = E_NUM / 16;   // 100000 exact
    for (int tile = blockIdx.x * 8 + wave; tile < numTiles; tile += gridDim.x * 8) {
        const int e0 = tile * 16;

        // edge indices: lanes 0-15 -> src rows, lanes 16-31 -> dst rows (same-wave DS is in-order)
        idxB[lane] = (int)eidx[(size_t)hi * E_NUM + e0 + col];

        // gather pair = [x[src] | x[dst]] into LDS, b128 granularity: 512 float4 chunks
        #pragma unroll
        for (int it = 0; it < 16; ++it) {
            const int c    = it * 32 + lane;
            const int row  = c >> 5;
            const int rem  = c & 31;
            const int half = rem >> 4;
            const int q    = rem & 15;
            const int node = idxB[half * 16 + row];
            const float4 v = *(const float4*)(x + (size_t)node * 64 + q * 4);
            *(float4*)(pairA + row * RS_PAIR + half * 64 + q * 4) = v;
        }
        // gather edge_attr tile (fully coalesced b128)
        #pragma unroll
        for (int it = 0; it < 8; ++it) {
            const int c   = it * 32 + lane;
            const int row = c >> 4;
            const int q   = c & 15;
            const float4 v = *(const float4*)(eattr + (size_t)(e0 + row) * 64 + q * 4);
            *(float4*)(attrA + row * RS_ATTR + q * 4) = v;
        }

        // ---------------- layer 1: hidden = relu(pair@Wnnn+b) + relu(attr@Wroot+b) ----------------
        // k-outer / ntile-inner: one A fragment feeds 8 WMMAs; B fragments at constant DS offsets.
        v8f h[NT], g[NT];
        #pragma unroll
        for (int nt = 0; nt < NT; ++nt) {
            const float bn = bb_n[nt];
            const float br = bb_r[nt];
            const v8f tn = {bn, bn, bn, bn, bn, bn, bn, bn};  // bias folded into accumulator
            const v8f tr = {br, br, br, br, br, br, br, br};
            h[nt] = tn;
            g[nt] = tr;
        }

        {   // pair @ Wnnn
            const float* ap = pairA + col * RS_PAIR + 2 * hi;
            const v2f*   bk = bW1;
            #pragma unroll 2
            for (int k = 0; k < KS1; ++k) {
                const v2f a = *(const v2f*)ap;
                #pragma unroll
                for (int nt = 0; nt < NT; ++nt)
                    h[nt] = __builtin_amdgcn_wmma_f32_16x16x4_f32(false, a, false, bk[nt * KS1 * 32],
                                                                  (short)0, h[nt], false, false);
                ap += 4;
                bk += 32;
            }
        }
        {   // attr @ Wroot
            const float* ap = attrA + col * RS_ATTR + 2 * hi;
            const v2f*   bk = bWR;
            #pragma unroll 2
            for (int k = 0; k < KSR; ++k) {
                const v2f a = *(const v2f*)ap;
                #pragma unroll
                for (int nt = 0; nt < NT; ++nt)
                    g[nt] = __builtin_amdgcn_wmma_f32_16x16x4_f32(false, a, false, bk[nt * KSR * 32],
                                                                  (short)0, g[nt], false, false);
                ap += 4;
                bk += 32;
            }
        }
        #pragma unroll
        for (int nt = 0; nt < NT; ++nt) {
            #pragma unroll
            for (int v = 0; v < 8; ++v)
                h[nt][v] = fmaxf(h[nt][v], 0.0f) + fmaxf(g[nt][v], 0.0f);
        }

        // C-layout -> A-layout reshape via LDS; pairA is fully consumed, safe to reuse (same-wave DS order)
        #pragma unroll
        for (int nt = 0; nt < NT; ++nt) {
            #pragma unroll
            for (int v = 0; v < 8; ++v)
                pairA[(v + 8 * hi) * RS_PAIR + nt * 16 + col] = h[nt][v];
        }

        // ---------------- layer 2: out = relu(hidden@Wout + b) ----------------
        v8f o[NT];
        #pragma unroll
        for (int nt = 0; nt < NT; ++nt) {
            const float bo = bb_o[nt];
            const v8f to = {bo, bo, bo, bo, bo, bo, bo, bo};
            o[nt] = to;
        }
        {
            const float* ap = pairA + col * RS_PAIR + 2 * hi;   // pairA now holds hidden
            const v2f*   bk = bW2;
            #pragma unroll 2
            for (int k = 0; k < KS1; ++k) {
                const v2f a = *(const v2f*)ap;
                #pragma unroll
                for (int nt = 0; nt < NT; ++nt)
                    o[nt] = __builtin_amdgcn_wmma_f32_16x16x4_f32(false, a, false, bk[nt * KS1 * 32],
                                                                  (short)0, o[nt], false, false);
                ap += 4;
                bk += 32;
            }
        }
        #pragma unroll
        for (int nt = 0; nt < NT; ++nt) {
            #pragma unroll
            for (int v = 0; v < 8; ++v)
                out[(size_t)(e0 + v + 8 * hi) * 128 + nt * 16 + col] = fmaxf(o[nt][v], 0.0f);
        }
    }
}

extern "C" void kernel_launch(void* const* d_in, const int* in_sizes, int n_in,
                              void* d_out, int out_size, void* d_ws, size_t ws_size,
                              hipStream_t stream) {
    (void)in_sizes; (void)n_in; (void)out_size; (void)d_ws; (void)ws_size;
    const float*     x     = (const float*)d_in[0];
    const long long* eidx  = (const long long*)d_in[1];
    const float*     eattr = (const float*)d_in[2];
    const float*     Wnnn  = (const float*)d_in[3];
    const float*     bnnn  = (const float*)d_in[4];
    const float*     Wroot = (const float*)d_in[5];
    const float*     broot = (const float*)d_in[6];
    const float*     Wout  = (const float*)d_in[7];
    const float*     bout  = (const float*)d_in[8];
    float*           out   = (float*)d_out;

    const size_t shmem_bytes = (size_t)TOTAL_DW * sizeof(float);   // 267264 B
    // persistent grid-stride: amortizes the one-time LDS weight packing per block
    edge_mlp_wmma_f32<<<1024, 256, shmem_bytes, stream>>>(
        x, eidx, eattr, Wnnn, bnnn, Wroot, broot, Wout, bout, out);
}